// GRU_Model_36481452212265
// MI455X (gfx1250) — compile-verified
//
#include <hip/hip_runtime.h>
#include <math.h>

// ---------------------------------------------------------------------------
// Problem constants (from reference): E nodes, F feat, HID hidden, T steps
// ---------------------------------------------------------------------------
constexpr int E_   = 2048;
constexpr int F_   = 256;
constexpr int HID_ = 128;
constexpr int T_   = 32;
constexpr int GD_  = 2 * HID_;   // 256, width of G / H / gates

// WMMA fragment vector types (fp32 path: V_WMMA_F32_16X16X4_F32)
typedef float v2f __attribute__((ext_vector_type(2)));
typedef float v8f __attribute__((ext_vector_type(8)));

// GEMM tiling: 256 threads = 8 wave32; macro tile 128(M) x 64(N),
// waves arranged 4x2, each wave computes 32x32 = 2x2 WMMA 16x16 tiles.
#define BM 128
#define BN 64
#define KB 32
// LDS strides (floats). Both 16B-aligned rows AND bank-conflict-free for the
// fragment read patterns (proof: 36=4*9, 9 coprime 16 -> distinct banks over
// 16 rows; 36*d==2 (mod 64) unsolvable; 2*72 mod 64 = 16 separates B halves).
#define LDA_S 36
#define LDB_S 72

// Epilogue modes
#define EPI_NONE          0
#define EPI_RELU          1
#define EPI_SIGMOID_BIAS  2
#define EPI_TANH_BIAS     3
#define EPI_AGTG          4   // v=|acc+b[row]|; v=max(v+adj*d,0); +1 on diag

// ---------------------------------------------------------------------------
// Tensor Data Mover staging (CDNA5). Descriptor packed per ISA 8.3/8.4.
// ---------------------------------------------------------------------------
#if __has_builtin(__builtin_amdgcn_tensor_load_to_lds) && \
    __has_builtin(__builtin_amdgcn_s_wait_tensorcnt)
#define HAVE_TDM 1
#else
#define HAVE_TDM 0
#endif

#if HAVE_TDM
typedef unsigned int u32x4 __attribute__((ext_vector_type(4)));
typedef int          i32x4 __attribute__((ext_vector_type(4)));
typedef int          i32x8 __attribute__((ext_vector_type(8)));

// 32-bit LDS byte offset of a __shared__ object (generic -> AS(3) -> int).
__device__ __forceinline__ unsigned lds_offset_u32(const void* p)
{
    return (unsigned)(unsigned long long)
        (const __attribute__((address_space(3))) void*)p;
}

// Issue one 2D TDM tile load: tile_w dwords x tile_h rows from a row-major
// fp32 matrix with row stride `row_stride` elems, into LDS at lds_byte_addr,
// inserting `pad` dwords of LDS padding every `interval` dwords so the LDS
// image lands with our conflict-free strides.
//   pad_int_code: interval = 2 << code     (A: 32 -> 4, B: 64 -> 5)
//   pad_amt_code: pad dwords = code + 1    (A:  4 -> 3, B:  8 -> 7)
__device__ __forceinline__ void tdm_load_2d(unsigned lds_byte_addr,
                                            const float* gptr,
                                            unsigned tile_w, unsigned tile_h,
                                            unsigned row_stride,
                                            unsigned pad_int_code,
                                            unsigned pad_amt_code)
{
    unsigned long long ga = (unsigned long long)gptr;

    u32x4 g0;
    g0[0] = 1u;                                            // count=1, user D#
    g0[1] = lds_byte_addr;                                 // lds_addr [63:32]
    g0[2] = (unsigned)ga;                                  // global_addr lo
    g0[3] = (unsigned)((ga >> 32) & 0x01FFFFFFu)           // global_addr[56:32]
          | (2u << 30);                                    // type=2 (image)

    const unsigned td0 = row_stride;                       // tensor_dim0
    const unsigned td1 = 1u << 20;                         // tensor_dim1 (big)
    const unsigned long long s0 = row_stride;              // dim0 stride

    i32x8 g1;
    g1[0] = (int)((2u << 16)                               // data_size = 4B
                | (1u << 20)                               // pad_enable
                | (pad_int_code << 22)
                | (pad_amt_code << 25));                   // wg_mask = 0
    g1[1] = (int)((td0 & 0xFFFFu) << 16);                  // tensor_dim0 lo16
    g1[2] = (int)(((td0 >> 16) & 0xFFFFu)                  // tensor_dim0 hi16
                | ((td1 & 0xFFFFu) << 16));                // tensor_dim1 lo16
    g1[3] = (int)(((td1 >> 16) & 0xFFFFu)                  // tensor_dim1 hi16
                | (tile_w << 16));                         // tile_dim0
    g1[4] = (int)(tile_h);                                 // tile_dim1, dim2=0
    g1[5] = (int)(unsigned)(s0 & 0xFFFFFFFFull);           // stride0 lo32
    g1[6] = (int)(unsigned)((s0 >> 32) & 0xFFFFu);         // stride0 hi16
    g1[7] = 0;                                             // stride1 (unused)

    i32x4 g2 = {0, 0, 0, 0};
    i32x4 g3 = {0, 0, 0, 0};
#if defined(__clang_major__) && (__clang_major__ >= 23)
    i32x8 g4 = {0, 0, 0, 0, 0, 0, 0, 0};
    __builtin_amdgcn_tensor_load_to_lds(g0, g1, g2, g3, g4, 0);
#else
    __builtin_amdgcn_tensor_load_to_lds(g0, g1, g2, g3, 0);
#endif
}
#endif // HAVE_TDM

// ---------------------------------------------------------------------------
// Fused fp32 WMMA GEMM:  C = epi( A0@B0 [+ A1@B1] [+bias] )
// A: row-major MxK (lda), B: row-major KxN (ldb), C: row-major (ldc).
// M,N,K multiples of BM/BN/KB respectively (true for all call sites here).
// ---------------------------------------------------------------------------
__global__ __launch_bounds__(256)
void wmma_gemm_f32(const float* __restrict__ A0, int lda0,
                   const float* __restrict__ B0, int ldb0, int K0,
                   const float* __restrict__ A1, int lda1,
                   const float* __restrict__ B1, int ldb1, int K1,
                   float* __restrict__ C, int ldc,
                   const float* __restrict__ bias,     // per-row, or null
                   const float* __restrict__ adj, int ldadj,
                   const float* __restrict__ dscalar,  // 1-elem, AGTG only
                   int epi)
{
    __shared__ __align__(16) float As[2][BM * LDA_S];
    __shared__ __align__(16) float Bs[2][KB * LDB_S];

    const int tid   = threadIdx.x;
    const int lane  = tid & 31;
    const int w     = tid >> 5;
    const int mW    = (w >> 1) * 32;     // wave row offset in macro tile
    const int nW    = (w & 1) * 32;      // wave col offset in macro tile
    const int hl    = lane >> 4;         // lane half (0: lanes 0-15, 1: 16-31)
    const int l16   = lane & 15;
    const int blockM = blockIdx.y * BM;
    const int blockN = blockIdx.x * BN;

    v8f acc[2][2];
    #pragma unroll
    for (int mi = 0; mi < 2; ++mi)
        #pragma unroll
        for (int ni = 0; ni < 2; ++ni)
            #pragma unroll
            for (int k = 0; k < 8; ++k)
                acc[mi][ni][k] = 0.0f;

    // 32 WMMA ops over one staged KB-slab pair.
    auto compute_slab = [&](const float* Asb, const float* Bsb) {
        #pragma unroll
        for (int kk = 0; kk < KB; kk += 4) {
            v2f a[2], b[2];
            const int kc = kk + (hl << 1);  // lanes 0-15: K0/1, 16-31: K2/3
            #pragma unroll
            for (int mi = 0; mi < 2; ++mi) {
                int row = mW + mi * 16 + l16;
                a[mi][0] = Asb[row * LDA_S + kc];
                a[mi][1] = Asb[row * LDA_S + kc + 1];
            }
            #pragma unroll
            for (int ni = 0; ni < 2; ++ni) {
                int col = nW + ni * 16 + l16;
                b[ni][0] = Bsb[kc * LDB_S + col];
                b[ni][1] = Bsb[(kc + 1) * LDB_S + col];
            }
            #pragma unroll
            for (int mi = 0; mi < 2; ++mi)
                #pragma unroll
                for (int ni = 0; ni < 2; ++ni)
                    acc[mi][ni] = __builtin_amdgcn_wmma_f32_16x16x4_f32(
                        false, a[mi], false, b[ni],
                        (short)0, acc[mi][ni], false, false);
        }
    };

    // Two accumulation passes: (A0,B0,K0) then optionally (A1,B1,K1).
    for (int pass = 0; pass < 2; ++pass) {
        const float* A  = pass ? A1 : A0;
        const float* B  = pass ? B1 : B0;
        const int   lda = pass ? lda1 : lda0;
        const int   ldb = pass ? ldb1 : ldb0;
        const int   K   = pass ? K1  : K0;
        if (A == nullptr) continue;
        const int ntiles = K / KB;

#if HAVE_TDM
        // ---- Tensor Data Mover staging, double buffered ----
        const unsigned offA0 = lds_offset_u32(&As[0][0]);
        const unsigned offA1 = lds_offset_u32(&As[1][0]);
        const unsigned offB0 = lds_offset_u32(&Bs[0][0]);
        const unsigned offB1 = lds_offset_u32(&Bs[1][0]);
        if (w == 0) {   // one wave issues DMA (TDM ignores EXEC, per-wave op)
            tdm_load_2d(offA0, A + (size_t)blockM * lda, KB, BM, lda, 4, 3);
            tdm_load_2d(offB0, B + blockN, BN, KB, ldb, 5, 7);
        }
        for (int i = 0; i < ntiles; ++i) {
            const int cur = i & 1;
            if (w == 0) {
                if (i + 1 < ntiles) {
                    const int k0n = (i + 1) * KB;
                    tdm_load_2d(cur ? offA0 : offA1,
                                A + (size_t)blockM * lda + k0n, KB, BM, lda, 4, 3);
                    tdm_load_2d(cur ? offB0 : offB1,
                                B + (size_t)k0n * ldb + blockN, BN, KB, ldb, 5, 7);
                    __builtin_amdgcn_s_wait_tensorcnt(2);  // tile i landed
                } else {
                    __builtin_amdgcn_s_wait_tensorcnt(0);
                }
            }
            __syncthreads();                // tile i visible to all waves
            compute_slab(&As[cur][0], &Bs[cur][0]);
            __syncthreads();                // all reads done before overwrite
        }
#else
        // ---- fallback: manual global->VGPR->LDS staging (buffer 0) ----
        for (int i = 0; i < ntiles; ++i) {
            const int k0 = i * KB;
            #pragma unroll
            for (int j = 0; j < 4; ++j) {   // A tile: 1024 float4
                int idx = tid + j * 256;    // 8 float4 per row
                int r   = idx >> 3;
                int c4  = (idx & 7) << 2;
                float4 v = *reinterpret_cast<const float4*>(
                    &A[(size_t)(blockM + r) * lda + k0 + c4]);
                *reinterpret_cast<float4*>(&As[0][r * LDA_S + c4]) = v;
            }
            #pragma unroll
            for (int j = 0; j < 2; ++j) {   // B tile: 512 float4
                int idx = tid + j * 256;    // 16 float4 per row
                int r   = idx >> 4;
                int c4  = (idx & 15) << 2;
                float4 v = *reinterpret_cast<const float4*>(
                    &B[(size_t)(k0 + r) * ldb + blockN + c4]);
                *reinterpret_cast<float4*>(&Bs[0][r * LDB_S + c4]) = v;
            }
            __syncthreads();
            compute_slab(&As[0][0], &Bs[0][0]);
            __syncthreads();
        }
#endif
    }

    // --- fused epilogue + store (C/D layout: vgpr j -> M = hl*8 + j) ---
    const float dval = (epi == EPI_AGTG) ? dscalar[0] : 0.0f;
    #pragma unroll
    for (int mi = 0; mi < 2; ++mi) {
        #pragma unroll
        for (int ni = 0; ni < 2; ++ni) {
            #pragma unroll
            for (int j = 0; j < 8; ++j) {
                int row = blockM + mW + mi * 16 + (hl << 3) + j;
                int col = blockN + nW + ni * 16 + l16;
                float v = acc[mi][ni][j];
                if (epi == EPI_RELU) {
                    v = fmaxf(v, 0.0f);
                } else if (epi == EPI_SIGMOID_BIAS) {
                    v = v + bias[row];
                    v = 1.0f / (1.0f + __expf(-v));
                } else if (epi == EPI_TANH_BIAS) {
                    v = tanhf(v + bias[row]);
                } else if (epi == EPI_AGTG) {
                    v = fabsf(v + bias[row]);
                    v = fmaxf(v + adj[(size_t)row * ldadj + col] * dval, 0.0f);
                    if (row == col) v += 1.0f;
                }
                C[(size_t)row * ldc + col] = v;
            }
        }
    }
}

// ---------------------------------------------------------------------------
// dinv[row] = rsqrt( sum_j A[row,j] )
// ---------------------------------------------------------------------------
__global__ __launch_bounds__(256)
void rowsum_rsqrt(const float* __restrict__ A, float* __restrict__ dinv, int n)
{
    __shared__ float red[256];
    const int row = blockIdx.x;
    const float* a = A + (size_t)row * n;
    float s = 0.0f;
    for (int j = threadIdx.x; j < n; j += 256) s += a[j];
    red[threadIdx.x] = s;
    __syncthreads();
    for (int off = 128; off > 0; off >>= 1) {
        if (threadIdx.x < off) red[threadIdx.x] += red[threadIdx.x + off];
        __syncthreads();
    }
    if (threadIdx.x == 0) dinv[row] = rsqrtf(red[0]);
}

// A[i,j] *= dinv[i]*dinv[j]   (in place, float4)
__global__ __launch_bounds__(256)
void norm_scale(float* __restrict__ A, const float* __restrict__ dinv, int n)
{
    const int idx4 = blockIdx.x * 256 + threadIdx.x;
    const int row  = idx4 / (n >> 2);
    const int c4   = (idx4 % (n >> 2)) << 2;
    const float di = dinv[row];
    float4* p = reinterpret_cast<float4*>(A + (size_t)row * n + c4);
    float4 v = *p;
    v.x *= di * dinv[c4 + 0];
    v.y *= di * dinv[c4 + 1];
    v.z *= di * dinv[c4 + 2];
    v.w *= di * dinv[c4 + 3];
    *p = v;
}

// ---------------------------------------------------------------------------
// Elementwise helpers (float4)
// ---------------------------------------------------------------------------
__global__ __launch_bounds__(256)
void ew_zero(float4* __restrict__ p, int n4)
{
    int i = blockIdx.x * 256 + threadIdx.x;
    if (i < n4) p[i] = make_float4(0.f, 0.f, 0.f, 0.f);
}

__global__ __launch_bounds__(256)
void ew_mul(const float4* __restrict__ a, const float4* __restrict__ b,
            float4* __restrict__ o, int n4)
{
    int i = blockIdx.x * 256 + threadIdx.x;
    if (i < n4) {
        float4 x = a[i], y = b[i];
        o[i] = make_float4(x.x * y.x, x.y * y.y, x.z * y.z, x.w * y.w);
    }
}

// H = U*H + (1-U)*C
__global__ __launch_bounds__(256)
void ew_gru(const float4* __restrict__ u, const float4* __restrict__ c,
            float4* __restrict__ h, int n4)
{
    int i = blockIdx.x * 256 + threadIdx.x;
    if (i < n4) {
        float4 uu = u[i], cc = c[i], hh = h[i];
        hh.x = uu.x * hh.x + (1.0f - uu.x) * cc.x;
        hh.y = uu.y * hh.y + (1.0f - uu.y) * cc.y;
        hh.z = uu.z * hh.z + (1.0f - uu.z) * cc.z;
        hh.w = uu.w * hh.w + (1.0f - uu.w) * cc.w;
        h[i] = hh;
    }
}

// ---------------------------------------------------------------------------
// Host-side orchestration (graph-capture safe: fixed launch sequence)
// ---------------------------------------------------------------------------
extern "C" void kernel_launch(void* const* d_in, const int* in_sizes, int n_in,
                              void* d_out, int out_size, void* d_ws, size_t ws_size,
                              hipStream_t stream)
{
    const float* X    = (const float*)d_in[0];   // [T,E,F]
    const float* adj  = (const float*)d_in[1];   // [E,E]
    const float* P    = (const float*)d_in[2];   // [E,E]
    const float* Q    = (const float*)d_in[3];   // [F,E]
    const float* b    = (const float*)d_in[4];   // [E,1]
    const float* d    = (const float*)d_in[5];   // [1]
    const float* W1   = (const float*)d_in[6];   // [F,HID]
    const float* W2   = (const float*)d_in[7];   // [HID,HID]
    const float* Wr_x = (const float*)d_in[8];   // [E,E]
    const float* Wr_h = (const float*)d_in[9];
    const float* br   = (const float*)d_in[10];
    const float* Wu_x = (const float*)d_in[11];
    const float* Wu_h = (const float*)d_in[12];
    const float* bu   = (const float*)d_in[13];
    const float* Wc_x = (const float*)d_in[14];
    const float* Wc_h = (const float*)d_in[15];
    const float* bc   = (const float*)d_in[16];

    float* out   = (float*)d_out;
    float* H     = out;                        // [E, 256]  (GRU state, output 0)
    float* Anorm = out + (size_t)E_ * GD_;     // [E, E]    (output 1; per-step scratch)

    float* ws   = (float*)d_ws;
    float* XP   = ws;  ws += (size_t)E_ * F_;    // P @ x_t          [E,F]
    float* XW1  = ws;  ws += (size_t)E_ * HID_;  // x_t @ W1         [E,HID]
    float* h1W2 = ws;  ws += (size_t)E_ * HID_;  // h1 @ W2          [E,HID]
    float* G    = ws;  ws += (size_t)E_ * GD_;   // [h1 | h2]        [E,256]
    float* R    = ws;  ws += (size_t)E_ * GD_;   // reset gate
    float* U    = ws;  ws += (size_t)E_ * GD_;   // update gate
    float* rH   = ws;  ws += (size_t)E_ * GD_;   // r * H
    float* Cc   = ws;  ws += (size_t)E_ * GD_;   // candidate
    float* dinv = ws;  ws += E_;                 // D^{-1/2}

    const dim3 blk(256);
    const int n4EG = (E_ * GD_) / 4;             // elementwise float4 count
    const dim3 ewg((n4EG + 255) / 256);

    // H starts at zero every call (deterministic across graph replays).
    ew_zero<<<ewg, blk, 0, stream>>>((float4*)H, n4EG);

    for (int t = 0; t < T_; ++t) {
        const float* Xt = X + (size_t)t * E_ * F_;

        // 1) XP = P @ x_t                         [2048,256] k=2048
        wmma_gemm_f32<<<dim3(F_ / BN, E_ / BM), blk, 0, stream>>>(
            P, E_, Xt, F_, E_, nullptr, 0, nullptr, 0, 0,
            XP, F_, nullptr, nullptr, 0, nullptr, EPI_NONE);

        // 2) A_I = agtg(XP @ Q)  -> Anorm region  [2048,2048] k=256
        wmma_gemm_f32<<<dim3(E_ / BN, E_ / BM), blk, 0, stream>>>(
            XP, F_, Q, E_, F_, nullptr, 0, nullptr, 0, 0,
            Anorm, E_, b, adj, E_, d, EPI_AGTG);

        // 3) dinv = rsqrt(rowsum(A_I));  4) A_norm = dinv[i]*A_I*dinv[j] in place
        rowsum_rsqrt<<<E_, blk, 0, stream>>>(Anorm, dinv, E_);
        norm_scale<<<(E_ * E_ / 4) / 256, blk, 0, stream>>>(Anorm, dinv, E_);

        // 5) XW1 = x_t @ W1                       [2048,128] k=256
        wmma_gemm_f32<<<dim3(HID_ / BN, E_ / BM), blk, 0, stream>>>(
            Xt, F_, W1, HID_, F_, nullptr, 0, nullptr, 0, 0,
            XW1, HID_, nullptr, nullptr, 0, nullptr, EPI_NONE);

        // 6) h1 = relu(A_norm @ XW1) -> G[:,0:128]
        wmma_gemm_f32<<<dim3(HID_ / BN, E_ / BM), blk, 0, stream>>>(
            Anorm, E_, XW1, HID_, E_, nullptr, 0, nullptr, 0, 0,
            G, GD_, nullptr, nullptr, 0, nullptr, EPI_RELU);

        // 7) h1W2 = G[:,0:128] @ W2               [2048,128] k=128
        wmma_gemm_f32<<<dim3(HID_ / BN, E_ / BM), blk, 0, stream>>>(
            G, GD_, W2, HID_, HID_, nullptr, 0, nullptr, 0, 0,
            h1W2, HID_, nullptr, nullptr, 0, nullptr, EPI_NONE);

        // 8) h2 = relu(A_norm @ h1W2) -> G[:,128:256]
        wmma_gemm_f32<<<dim3(HID_ / BN, E_ / BM), blk, 0, stream>>>(
            Anorm, E_, h1W2, HID_, E_, nullptr, 0, nullptr, 0, 0,
            G + HID_, GD_, nullptr, nullptr, 0, nullptr, EPI_RELU);

        // 9) r = sigmoid(Wr_x@G + Wr_h@H + br)    dual-K fused
        wmma_gemm_f32<<<dim3(GD_ / BN, E_ / BM), blk, 0, stream>>>(
            Wr_x, E_, G, GD_, E_, Wr_h, E_, H, GD_, E_,
            R, GD_, br, nullptr, 0, nullptr, EPI_SIGMOID_BIAS);

        // 10) u = sigmoid(Wu_x@G + Wu_h@H + bu)
        wmma_gemm_f32<<<dim3(GD_ / BN, E_ / BM), blk, 0, stream>>>(
            Wu_x, E_, G, GD_, E_, Wu_h, E_, H, GD_, E_,
            U, GD_, bu, nullptr, 0, nullptr, EPI_SIGMOID_BIAS);

        // 11) rH = r * H
        ew_mul<<<ewg, blk, 0, stream>>>((const float4*)R, (const float4*)H,
                                        (float4*)rH, n4EG);

        // 12) c = tanh(Wc_x@G + Wc_h@rH + bc)
        wmma_gemm_f32<<<dim3(GD_ / BN, E_ / BM), blk, 0, stream>>>(
            Wc_x, E_, G, GD_, E_, Wc_h, E_, rH, GD_, E_,
            Cc, GD_, bc, nullptr, 0, nullptr, EPI_TANH_BIAS);

        // 13) H = u*H + (1-u)*c
        ew_gru<<<ewg, blk, 0, stream>>>((const float4*)U, (const float4*)Cc,
                                        (float4*)H, n4EG);
    }
    // Outputs: H in out[0 : E*256], last A_norm in out[E*256 : E*256+E*E].
}